// RoIAlignRotated_23235773071465
// MI455X (gfx1250) — compile-verified
//
#include <hip/hip_runtime.h>
#include <cmath>

// Problem constants (from reference setup_inputs)
#define OUTD  7
#define NB    4
#define CCH   256
#define HH    200
#define WW    176
#define NROI  512
#define HWSZ  (HH*WW)        // 35200
#define CHW   (CCH*HWSZ)     // 9,011,200

typedef __attribute__((ext_vector_type(2))) float v2f;
typedef __attribute__((ext_vector_type(8))) float v8f;

#if defined(__gfx1250__) && __has_builtin(__builtin_amdgcn_wmma_f32_16x16x4_f32)
#define USE_WMMA 1
#else
#define USE_WMMA 0
#endif

// One block per (roi, bin). 256 threads = 8 wave32.
// Lanes 0..3 compute the 4 subsample points' metadata once into LDS.
// WMMA path: each wave accumulates two 16-channel tiles via 4 chained
// V_WMMA_F32_16X16X4_F32 ops (K=4 corners per subsample), weights broadcast
// along N so every D column holds the bin's result for its channel row.
__global__ __launch_bounds__(256) void roi_align_rotated_kernel(
    const float* __restrict__ feats,   // (4,256,200,176) fp32
    const float* __restrict__ rois,    // (512,6) fp32
    float* __restrict__ out)           // (512,256,7,7) fp32
{
    __shared__ int   s_off[4];     // b*CHW + y0*W + x0 per subsample
    __shared__ float s_w[4][4];    // {w00,w01,w10,w11} * (valid ? 0.25 : 0)

    const int bid = blockIdx.x;
    const int r   = bid / (OUTD*OUTD);
    const int bin = bid % (OUTD*OUTD);
    const int ph  = bin / OUTD;
    const int pw  = bin % OUTD;
    const int tid = threadIdx.x;

    if (tid < 4) {
        const int s = tid, j = s >> 1, k = s & 1;     // j -> y subsample, k -> x
        const float* rp = rois + (size_t)r * 6;
        const int   b  = (int)rp[0];
        const float cx = (rp[1] + 0.0f)  * 2.0f;      // (x + X_OFF) * SCALE
        const float cy = (rp[2] + 40.0f) * 2.0f;      // (y + Y_OFF) * SCALE
        const float rw = fmaxf(rp[3] * 2.0f, 1.0f);
        const float rh = fmaxf(rp[4] * 2.0f, 1.0f);
        const float th = rp[5];
        const float bw = rw * (1.0f / OUTD);
        const float bh = rh * (1.0f / OUTD);
        const float yy = -0.5f * rh + bh * ((float)ph + ((float)j + 0.5f) * 0.5f);
        const float xx = -0.5f * rw + bw * ((float)pw + ((float)k + 0.5f) * 0.5f);
        const float ct = cosf(th), st = sinf(th);
        float x = xx * ct - yy * st + cx;
        float y = xx * st + yy * ct + cy;
        const bool valid = (y > -1.0f) && (y < (float)HH) &&
                           (x > -1.0f) && (x < (float)WW);
        y = fminf(fmaxf(y, 0.0f), (float)(HH - 1));
        x = fminf(fmaxf(x, 0.0f), (float)(WW - 1));
        int y0 = (int)floorf(y); if (y0 > HH - 2) y0 = HH - 2;
        int x0 = (int)floorf(x); if (x0 > WW - 2) x0 = WW - 2;
        const float ly = y - (float)y0, lx = x - (float)x0;
        const float hy = 1.0f - ly,     hx = 1.0f - lx;
        const float ws = valid ? 0.25f : 0.0f;        // fold mean over 2x2 samples
        s_off[s]  = b * CHW + y0 * WW + x0;
        s_w[s][0] = hy * hx * ws;
        s_w[s][1] = hy * lx * ws;
        s_w[s][2] = ly * hx * ws;
        s_w[s][3] = ly * lx * ws;
    }
    __syncthreads();

#if USE_WMMA
    const int  lane = tid & 31;
    const int  wv   = tid >> 5;
    const bool hi   = lane >= 16;         // lane half selects K={0,1} vs K={2,3}
    const int  m    = lane & 15;          // A-matrix row (channel within tile)
    const int  rowoff = hi ? WW : 0;      // high lanes sample the y1 row
    const int  cA = wv * 32 + m;          // tile A channels
    const int  cB = cA + 16;              // tile B channels

    v8f accA = {};
    v8f accB = {};

    #pragma unroll
    for (int s = 0; s < 4; ++s) {
        const int base = s_off[s] + rowoff;
        // f32 A 16x4 layout (ISA 7.12.2): lanes0-15 VGPR0/1 = K0/K1 (v00,v01),
        // lanes16-31 VGPR0/1 = K2/K3 (v10,v11). Adjacent-address pair per lane.
        const float* pA = feats + (size_t)base + (size_t)cA * HWSZ;
        const float* pB = feats + (size_t)base + (size_t)cB * HWSZ;
        v2f a;  a.x  = pA[0]; a.y  = pA[1];
        v2f bb; bb.x = pB[0]; bb.y = pB[1];
        // f32 B 4x16: weights broadcast along N; K rows split by lane half
        // (inferred from A/C/D lane-half pattern — only undocumented layout).
        v2f w;  w.x = hi ? s_w[s][2] : s_w[s][0];
                w.y = hi ? s_w[s][3] : s_w[s][1];
        accA = __builtin_amdgcn_wmma_f32_16x16x4_f32(
                   false, a,  false, w, (short)0, accA, false, false);
        accB = __builtin_amdgcn_wmma_f32_16x16x4_f32(
                   false, bb, false, w, (short)0, accB, false, false);
    }

    // D 16x16 f32 layout: VGPR v -> (M = v + 8*hi, N = lane&15). All N columns
    // identical by construction; lanes with N==0 in each half write out.
    if (m == 0) {
        const int half = hi ? 8 : 0;
        float* op = out + (size_t)r * CCH * (OUTD*OUTD) + bin;
        const int chA = wv * 32 + half;
        const int chB = chA + 16;
        #pragma unroll
        for (int v = 0; v < 8; ++v) {
            op[(size_t)(chA + v) * (OUTD*OUTD)] = accA[v];
            op[(size_t)(chB + v) * (OUTD*OUTD)] = accB[v];
        }
    }
#else
    // VALU fallback: one thread per channel.
    const int c = tid;
    const float* pc = feats + (size_t)c * HWSZ;
    float acc = 0.0f;
    #pragma unroll
    for (int s = 0; s < 4; ++s) {
        const int base = s_off[s];
        const float f00 = pc[base];
        const float f01 = pc[base + 1];
        const float f10 = pc[base + WW];
        const float f11 = pc[base + WW + 1];
        acc += s_w[s][0] * f00 + s_w[s][1] * f01 +
               s_w[s][2] * f10 + s_w[s][3] * f11;
    }
    out[((size_t)r * CCH + c) * (OUTD*OUTD) + bin] = acc;
#endif
}

extern "C" void kernel_launch(void* const* d_in, const int* in_sizes, int n_in,
                              void* d_out, int out_size, void* d_ws, size_t ws_size,
                              hipStream_t stream) {
    const float* feats = (const float*)d_in[0];   // (4,256,200,176) fp32
    const float* rois  = (const float*)d_in[1];   // (512,6) fp32
    float* out = (float*)d_out;                   // (512,256,7,7) fp32
    (void)in_sizes; (void)n_in; (void)out_size; (void)d_ws; (void)ws_size;

    const int nblocks = NROI * OUTD * OUTD;       // 25,088 blocks, one per (roi,bin)
    roi_align_rotated_kernel<<<nblocks, 256, 0, stream>>>(feats, rois, out);
}